// MEMC_Net_79388175499807
// MI455X (gfx1250) — compile-verified
//
#include <hip/hip_runtime.h>
#include <math.h>

// Problem constants (match reference)
constexpr int Bn = 4, Cn = 3, Hn = 480, Wn = 854;
constexpr int FSZ = 4;                 // filter size
constexpr int HW = Hn * Wn;            // plane stride
constexpr int NPIX = Bn * HW;          // 1,639,680 (= 6405 * 256 exactly)

// One flow-warped adaptive-filter sample for all 3 channels of one pixel.
// img  : base of [C,H,W] planes for this batch
// off  : base of [2,H,W] flow for this batch
// filt : base of [16,H,W] filter for this batch
__device__ __forceinline__ void filter_warp(
    const float* __restrict__ img,
    const float* __restrict__ off,
    const float* __restrict__ filt,
    int h, int w, int idx_hw, float acc[Cn])
{
    const float fx = off[idx_hw];
    const float fy = off[HW + idx_hw];
    const float x2 = (float)w + fx;
    const float y2 = (float)h + fy;

    // Reference multiplies the result by the valid mask -> zero contribution.
    if (!(x2 >= 0.0f && x2 <= (float)(Wn - 1) &&
          y2 >= 0.0f && y2 <= (float)(Hn - 1))) {
        acc[0] = acc[1] = acc[2] = 0.0f;
        return;
    }

    const int   ix = (int)floorf(x2);
    const int   iy = (int)floorf(y2);
    const float a  = x2 - (float)ix;       // bilinear frac x (alpha)
    const float bb = y2 - (float)iy;       // bilinear frac y (beta)
    const float na = 1.0f - a;
    const float nb = 1.0f - bb;

    // Load the 16 per-pixel filter taps (plane-strided, lane-coalesced).
    float f[FSZ * FSZ];
#pragma unroll
    for (int k = 0; k < FSZ * FSZ; ++k)
        f[k] = filt[k * HW + idx_hw];

    // Fold the 2x2 bilinear blend into the filter (separable):
    //   out = sum_{J,I in 5x5} Wt[J][I] * win[J][I]
    // with Wt = f (4x4) convolved by [[nb*na, nb*a],[bb*na, bb*a]].
    float Hr[FSZ][FSZ + 1];                // horizontal pass: 4x5
#pragma unroll
    for (int j = 0; j < FSZ; ++j) {
        Hr[j][0] = na * f[j * FSZ + 0];
#pragma unroll
        for (int i = 1; i < FSZ; ++i)
            Hr[j][i] = fmaf(na, f[j * FSZ + i], a * f[j * FSZ + i - 1]);
        Hr[j][FSZ] = a * f[j * FSZ + FSZ - 1];
    }
    float Wt[FSZ + 1][FSZ + 1];            // vertical pass: 5x5
#pragma unroll
    for (int i = 0; i < FSZ + 1; ++i) {
        Wt[0][i] = nb * Hr[0][i];
#pragma unroll
        for (int j = 1; j < FSZ; ++j)
            Wt[j][i] = fmaf(nb, Hr[j][i], bb * Hr[j - 1][i]);
        Wt[FSZ][i] = bb * Hr[FSZ - 1][i];
    }

    // Clamped window coordinates (shared by all 3 channels).
    int xoff[FSZ + 1], yoff[FSZ + 1];
#pragma unroll
    for (int t = 0; t < FSZ + 1; ++t) {
        int xc = ix - 1 + t;  xc = xc < 0 ? 0 : (xc > Wn - 1 ? Wn - 1 : xc);
        int yc = iy - 1 + t;  yc = yc < 0 ? 0 : (yc > Hn - 1 ? Hn - 1 : yc);
        xoff[t] = xc;
        yoff[t] = yc * Wn;
    }

    // 25 gathers + 25 FMAs per channel (instead of 4x16 dot products).
#pragma unroll
    for (int c = 0; c < Cn; ++c) {
        const float* __restrict__ plane = img + c * HW;
        float s = 0.0f;
#pragma unroll
        for (int J = 0; J < FSZ + 1; ++J) {
            const float* __restrict__ row = plane + yoff[J];
#pragma unroll
            for (int I = 0; I < FSZ + 1; ++I)
                s = fmaf(Wt[J][I], row[xoff[I]], s);
        }
        acc[c] = s;
    }
}

__global__ __launch_bounds__(256) void memc_filter_interp_kernel(
    const float* __restrict__ ref0, const float* __restrict__ ref2,
    const float* __restrict__ off0, const float* __restrict__ off1,
    const float* __restrict__ fil0, const float* __restrict__ fil1,
    const float* __restrict__ occ0, const float* __restrict__ occ1,
    float* __restrict__ out)
{
    const int idx = blockIdx.x * blockDim.x + threadIdx.x;
    if (idx >= NPIX) return;

    const int b      = idx / HW;
    const int idx_hw = idx - b * HW;
    const int h      = idx_hw / Wn;
    const int w      = idx_hw - h * Wn;

    const float* f0b = fil0 + b * (FSZ * FSZ) * HW;
    const float* f1b = fil1 + b * (FSZ * FSZ) * HW;

    // gfx1250 prefetch path (global_prefetch_b8): pre-touch the dominant
    // streaming operand (filter planes) ~48KB ahead of the current front.
    constexpr int PFD = 12288;  // elements ahead
    if (idx_hw + PFD < HW) {
        __builtin_prefetch(f0b + idx_hw + PFD, 0, 0);
        __builtin_prefetch(f1b + idx_hw + PFD, 0, 0);
    }

    float a0[Cn], a1[Cn];
    filter_warp(ref0 + b * Cn * HW, off0 + b * 2 * HW, f0b, h, w, idx_hw, a0);
    filter_warp(ref2 + b * Cn * HW, off1 + b * 2 * HW, f1b, h, w, idx_hw, a1);

    const float o0 = occ0[idx];
    const float o1 = occ1[idx];

#pragma unroll
    for (int c = 0; c < Cn; ++c)
        out[(b * Cn + c) * HW + idx_hw] = fmaf(o0, a0[c], o1 * a1[c]);
}

extern "C" void kernel_launch(void* const* d_in, const int* in_sizes, int n_in,
                              void* d_out, int out_size, void* d_ws, size_t ws_size,
                              hipStream_t stream)
{
    (void)in_sizes; (void)n_in; (void)out_size; (void)d_ws; (void)ws_size;

    const float* ref0 = (const float*)d_in[0];
    const float* ref2 = (const float*)d_in[1];
    const float* off0 = (const float*)d_in[2];
    const float* off1 = (const float*)d_in[3];
    const float* fil0 = (const float*)d_in[4];
    const float* fil1 = (const float*)d_in[5];
    const float* occ0 = (const float*)d_in[6];
    const float* occ1 = (const float*)d_in[7];
    float* out = (float*)d_out;

    const int threads = 256;                      // 8 wave32 waves / block
    const int blocks  = (NPIX + threads - 1) / threads;  // 6405, exact
    memc_filter_interp_kernel<<<blocks, threads, 0, stream>>>(
        ref0, ref2, off0, off1, fil0, fil1, occ0, occ1, out);
}